// GCNConv_69028714381389
// MI455X (gfx1250) — compile-verified
//
#include <hip/hip_runtime.h>

typedef __attribute__((ext_vector_type(2))) float v2f;
typedef __attribute__((ext_vector_type(8))) float v8f;

#define GCN_NIN  128
#define GCN_NOUT 128

// ---------------------------------------------------------------------------
// K0: deg[i] = 1.0f (self loop contribution to destination degree)
// ---------------------------------------------------------------------------
__global__ void k_deg_init(float* __restrict__ deg, int n) {
  int i = blockIdx.x * blockDim.x + threadIdx.x;
  if (i < n) deg[i] = 1.0f;
}

// ---------------------------------------------------------------------------
// K1: deg[dst[e]] += 1  over all edges
// ---------------------------------------------------------------------------
__global__ void k_deg_accum(const long long* __restrict__ dst,
                            float* __restrict__ deg, int e) {
  int i = blockIdx.x * blockDim.x + threadIdx.x;
  if (i < e) atomicAdd(&deg[(int)dst[i]], 1.0f);
}

// ---------------------------------------------------------------------------
// K2: dinv[i] = rsqrt(deg[i])   (deg >= 1 always, self loops)
// ---------------------------------------------------------------------------
__global__ void k_deg_rsqrt(float* __restrict__ deg, int n) {
  int i = blockIdx.x * blockDim.x + threadIdx.x;
  if (i < n) deg[i] = __frsqrt_rn(deg[i]);
}

// ---------------------------------------------------------------------------
// K3: h = x @ W^T  via V_WMMA_F32_16X16X4_F32 (full fp32 precision).
// Block = 256 threads = 8 waves. Wave w owns the 16x16 tile at
// (m0 = blockIdx.x*16, n0 = w*16). K walked in steps of 4, fully unrolled.
//
// A (16x4 f32, 2 VGPRs): lane L<16 -> (x[m0+L][k], x[m0+L][k+1]);
//                        lane L>=16 -> same row, k+2/k+3.
// B (4x16 f32, 2 VGPRs): B[k][o] = W[o][k]; lane L<16 -> (W[n0+L][k], W[n0+L][k+1]);
//                        lanes 16-31 -> k+2/k+3.  (row-major W[out][in])
// D (16x16 f32, 8 VGPRs): element r -> row m0 + r + 8*(lane>=16), col n0+(lane&15).
// ---------------------------------------------------------------------------
__global__ void __launch_bounds__(256)
k_gemm_wmma(const float* __restrict__ x, const float* __restrict__ W,
            float* __restrict__ h) {
  const int wave = threadIdx.x >> 5;
  const int lane = threadIdx.x & 31;
  const int m0 = blockIdx.x * 16;
  const int n0 = wave * 16;
  const int l15 = lane & 15;
  const int khi = (lane >> 4) * 2;                  // 0 or 2

  const float* __restrict__ xp = x + (m0 + l15) * GCN_NIN + khi;
  const float* __restrict__ wp = W + (n0 + l15) * GCN_NIN + khi;

  v8f c = {};
#pragma unroll
  for (int k = 0; k < GCN_NIN; k += 4) {
    v2f a = *(const v2f*)(xp + k);
    v2f b = *(const v2f*)(wp + k);
    c = __builtin_amdgcn_wmma_f32_16x16x4_f32(
        /*neg_a=*/false, a, /*neg_b=*/false, b,
        /*c_mod=*/(short)0, c, /*reuse_a=*/false, /*reuse_b=*/false);
  }

  const int rbase = m0 + (lane >> 4) * 8;
  float* __restrict__ hp = h + n0 + l15;
#pragma unroll
  for (int r = 0; r < 8; ++r) {
    hp[(long)(rbase + r) * GCN_NOUT] = c[r];
  }
}

// ---------------------------------------------------------------------------
// K4: out[i][c] = b[c] + h[i][c] * dinv[i]^2   (bias + self-loop message)
// float4 vectorized: 32 lanes cover one 128-wide row.
// ---------------------------------------------------------------------------
__global__ void k_out_init(const float* __restrict__ h,
                           const float* __restrict__ dinv,
                           const float* __restrict__ bias,
                           float* __restrict__ out, int n) {
  long i = (long)blockIdx.x * blockDim.x + threadIdx.x;   // over n*32 float4s
  if (i >= (long)n * (GCN_NOUT / 4)) return;
  int node = (int)(i >> 5);
  int c4 = (int)(i & 31);
  float di = dinv[node];
  float s = di * di;
  float4 hv = ((const float4*)h)[i];
  float4 bv = ((const float4*)bias)[c4];
  float4 o;
  o.x = bv.x + hv.x * s;
  o.y = bv.y + hv.y * s;
  o.z = bv.z + hv.z * s;
  o.w = bv.w + hv.w * s;
  ((float4*)out)[i] = o;
}

// ---------------------------------------------------------------------------
// K5: scatter  out[dst] += dinv[src]*dinv[dst] * h[src]
// One wave32 per edge; lane owns 4 contiguous channels (b128 gather,
// 4x global_atomic_add_f32; h/out are L2-resident at these sizes).
// ---------------------------------------------------------------------------
__global__ void __launch_bounds__(256)
k_scatter(const long long* __restrict__ src, const long long* __restrict__ dst,
          const float* __restrict__ h, const float* __restrict__ dinv,
          float* __restrict__ out, int e) {
  long gt = (long)blockIdx.x * blockDim.x + threadIdx.x;
  int edge = (int)(gt >> 5);
  int lane = (int)(gt & 31);
  if (edge >= e) return;
  int s = (int)src[edge];
  int d = (int)dst[edge];
  float norm = dinv[s] * dinv[d];
  float4 hv = ((const float4*)(h + (long)s * GCN_NOUT))[lane];
  float* op = out + (long)d * GCN_NOUT + lane * 4;
  atomicAdd(op + 0, hv.x * norm);
  atomicAdd(op + 1, hv.y * norm);
  atomicAdd(op + 2, hv.z * norm);
  atomicAdd(op + 3, hv.w * norm);
}

// ---------------------------------------------------------------------------
extern "C" void kernel_launch(void* const* d_in, const int* in_sizes, int n_in,
                              void* d_out, int out_size, void* d_ws, size_t ws_size,
                              hipStream_t stream) {
  const float* x = (const float*)d_in[0];
  const long long* edge_index = (const long long*)d_in[1];   // int64 [2, E]
  const float* W = (const float*)d_in[3];                    // [NOUT, NIN]
  const float* bias = (const float*)d_in[4];                 // [NOUT]

  const int n = in_sizes[0] / GCN_NIN;       // 100000
  const int e = in_sizes[1] / 2;             // 1600000
  const long long* srcp = edge_index;
  const long long* dstp = edge_index + e;

  float* out = (float*)d_out;
  float* h = (float*)d_ws;                   // n*128 floats (51.2 MB)
  float* dinv = h + (long)n * GCN_NOUT;      // n floats

  const int T = 256;

  // degrees -> dinv
  k_deg_init<<<(n + T - 1) / T, T, 0, stream>>>(dinv, n);
  k_deg_accum<<<(e + T - 1) / T, T, 0, stream>>>(dstp, dinv, e);
  k_deg_rsqrt<<<(n + T - 1) / T, T, 0, stream>>>(dinv, n);

  // dense transform h = x W^T (WMMA fp32); n is a multiple of 16
  k_gemm_wmma<<<n / 16, T, 0, stream>>>(x, W, h);

  // out = bias + self-loop message
  {
    long items = (long)n * (GCN_NOUT / 4);
    k_out_init<<<(unsigned)((items + T - 1) / T), T, 0, stream>>>(h, dinv, bias, out, n);
  }

  // edge scatter
  {
    long threads = (long)e * 32;
    k_scatter<<<(unsigned)((threads + T - 1) / T), T, 0, stream>>>(
        srcp, dstp, h, dinv, out, e);
  }
}